// YoloLoss_34617436406052
// MI455X (gfx1250) — compile-verified
//
#include <hip/hip_runtime.h>
#include <math.h>

// Problem constants (match reference)
#define Bv 32
#define Wv 64
#define Hv 64
#define Av 5
#define Cv 80
#define Mv 50
#define NCELL (Bv*Wv*Hv)          // 131072 grid cells
#define NCA   (Bv*Wv*Hv*Av)       // 655360 cell-anchors
#define NBM   (Bv*Mv)             // 1600 gt boxes
#define NPA   (NCA/256)           // 2560 blocks in dense pass

typedef float v2f __attribute__((ext_vector_type(2)));
typedef float v8f __attribute__((ext_vector_type(8)));

// ---- wave32 full reduction via V_WMMA_F32_16X16X4_F32 ----
// A vgpr0 = p  -> A[m][0] = p_m (lanes 0-15), A[m][2] = p_{m+16} (lanes 16-31)
// A vgpr1 = 0  -> K=1,3 are zero.  B = all ones (layout-independent).
// D[m][n] = sum_k A[m][k] = p_m + p_{m+16}; D vgpr v holds M=v (lanes 0-15)
// and M=v+8 (lanes 16-31). Sum of 8 D vgprs + xor-16 shuffle = full wave sum.
// Requires EXEC == all ones (call only from wave-uniform control flow).
__device__ __forceinline__ float wave_sum(float p) {
  v2f a; a[0] = p;   a[1] = 0.0f;
  v2f b; b[0] = 1.0f; b[1] = 1.0f;
  v8f c = {};
  v8f d = __builtin_amdgcn_wmma_f32_16x16x4_f32(
      false, a, false, b, (short)0, c, false, false);
  float s = d[0] + d[1] + d[2] + d[3] + d[4] + d[5] + d[6] + d[7];
  s += __shfl_xor(s, 16, 32);
  return s;
}

__device__ __forceinline__ float sigm(float x) {
  return 1.0f / (1.0f + expf(-x));
}

// ---- 1) init winner map ----
__global__ void kInit(int* __restrict__ winner) {
  int t = blockIdx.x * 256 + threadIdx.x;
  if (t < NCELL) winner[t] = -1;
}

// ---- 2) scatter: which gt box owns each cell (largest m wins) ----
__global__ void kScatter(const float* __restrict__ tobj, int* __restrict__ winner) {
  int t = blockIdx.x * 256 + threadIdx.x;
  if (t >= NBM) return;
  int b = t / Mv;
  const float* tr = tobj + t * 4;
  int ci = (int)floorf(tr[0]); ci = min(max(ci, 0), Wv - 1);
  int cj = (int)floorf(tr[1]); cj = min(max(cj, 0), Hv - 1);
  atomicMax(&winner[(b * Wv + ci) * Hv + cj], t % Mv);
}

// ---- 3) dense pass: sum conf^2 over all cell-anchors (13 MB stream) ----
__global__ void kConfSum(const float* __restrict__ pred, float* __restrict__ partial) {
  int t = blockIdx.x * 256 + threadIdx.x;     // exactly NCA threads
  float conf = sigm(pred[t * 5 + 4]);
  float s = wave_sum(conf * conf);            // EXEC all ones: full grid, no guards
  __shared__ float lds[8];
  int lane = threadIdx.x & 31, wv = threadIdx.x >> 5;
  if (lane == 0) lds[wv] = s;
  __syncthreads();
  if (threadIdx.x == 0) {
    float r = 0.f;
    for (int i = 0; i < 8; ++i) r += lds[i];
    partial[blockIdx.x] = r;
  }
}

// ---- 4) sparse pass: one wave per (b,m) gt box ----
__global__ void kObj(const float* __restrict__ cls, const float* __restrict__ pred,
                     const float* __restrict__ tlabel, const float* __restrict__ tobj,
                     const float* __restrict__ anch, const int* __restrict__ winner,
                     float* __restrict__ corrA, float* __restrict__ objA,
                     float* __restrict__ priorA, float* __restrict__ truA,
                     float* __restrict__ scoA) {
  int wid  = blockIdx.x * 8 + (threadIdx.x >> 5);  // 200 blocks * 8 waves = 1600 exactly
  int lane = threadIdx.x & 31;
  int b = wid / Mv, m = wid % Mv;
  const float* tr = tobj + wid * 4;
  float tx = tr[0], ty = tr[1], tw = tr[2], th = tr[3];
  int ci = (int)floorf(tx); ci = min(max(ci, 0), Wv - 1);
  int cj = (int)floorf(ty); cj = min(max(cj, 0), Hv - 1);

  float corr = 0.f, obj = 0.f, pri = 0.f, tru = 0.f, sco = 0.f;
  if (winner[(b * Wv + ci) * Hv + cj] == m) {          // wave-uniform branch
    for (int a = 0; a < Av; ++a) {
      float aw = anch[2 * a], ah = anch[2 * a + 1];
      float inter = fminf(tw, aw) * fminf(th, ah);
      float uni   = tw * th + aw * ah - inter;
      float iou   = inter / uni;
      if (iou > 0.5f) {                                // wave-uniform branch
        int cell = ((b * Wv + ci) * Hv + cj) * Av + a;
        const float* p = pred + (long)cell * 5;
        float conf = sigm(p[4]);
        corr += conf * conf;                           // obj-cell conf^2 (noobj correction)
        obj  += (conf - iou) * (conf - iou);
        float wx = expf(p[2]) * aw, wy = expf(p[3]) * ah;
        pri  += (wx - aw) * (wx - aw) + (wy - ah) * (wy - ah);
        float px = sigm(p[0]) + (float)ci;
        float py = sigm(p[1]) + (float)cj;
        tru  += (px - tx) * (px - tx) + (py - ty) * (py - ty)
              + (wx - tw) * (wx - tw) + (wy - th) * (wy - th);
        const float* srow = cls + (long)cell * Cv;     // 320 B coalesced gather
        const float* lrow = tlabel + (long)wid * Cv;
        for (int cc = lane; cc < Cv; cc += 32) {       // lane-split, reconverges
          float d = srow[cc] - lrow[cc];
          sco += d * d;
        }
      }
    }
  }
  sco = wave_sum(sco);                                 // EXEC all ones here
  if (lane == 0) {
    corrA[wid] = corr; objA[wid] = obj; priorA[wid] = pri;
    truA[wid]  = tru;  scoA[wid] = sco;
  }
}

// ---- 5) finalize: deterministic combine ----
__global__ void kFinal(const float* __restrict__ partialA, const float* __restrict__ corrA,
                       const float* __restrict__ objA, const float* __restrict__ priorA,
                       const float* __restrict__ truA, const float* __restrict__ scoA,
                       const int* __restrict__ iter, float* __restrict__ out) {
  int tid = threadIdx.x;
  float sA = 0.f;
  for (int i = tid; i < NPA; i += 256) sA += partialA[i];
  float sc = 0.f, so = 0.f, sp = 0.f, st = 0.f, ss = 0.f;
  for (int i = tid; i < NBM; i += 256) {
    sc += corrA[i]; so += objA[i]; sp += priorA[i]; st += truA[i]; ss += scoA[i];
  }
  __shared__ float lds[8];
  __shared__ float tot[6];
  float vals[6] = {sA, sc, so, sp, st, ss};
  int lane = tid & 31, wv = tid >> 5;
  for (int k = 0; k < 6; ++k) {
    float s = wave_sum(vals[k]);
    if (lane == 0) lds[wv] = s;
    __syncthreads();
    if (tid == 0) {
      float r = 0.f;
      for (int i = 0; i < 8; ++i) r += lds[i];
      tot[k] = r;
    }
    __syncthreads();
  }
  if (tid == 0) {
    const float N1 = (float)NCA;                     // 655360
    float noobj = 0.5f * 0.5f * (tot[0] - tot[1]) / N1;        // SCALE_NOOBJ=0.5
    float objl  = 0.5f * 5.0f * tot[2] / N1;                   // SCALE_OBJ=5.0
    float need  = (iter[0] < 12800) ? 1.f : 0.f;
    float prior = need * 0.5f * 5.0f * tot[3] / (2.0f * N1);
    float trul  = 0.5f * 5.0f * tot[4] / (4.0f * N1);
    float scol  = 0.5f * 5.0f * tot[5] / (80.0f * N1);
    out[0] = (noobj + objl + prior + trul + scol) * 0.25f;
  }
}

extern "C" void kernel_launch(void* const* d_in, const int* in_sizes, int n_in,
                              void* d_out, int out_size, void* d_ws, size_t ws_size,
                              hipStream_t stream) {
  const int*   iter  = (const int*)d_in[0];
  const float* cls   = (const float*)d_in[1];   // (B,W,H,A,C)
  const float* pred  = (const float*)d_in[2];   // (B,W,H,A,5)
  const float* tlab  = (const float*)d_in[3];   // (B,M,C)
  const float* tobj  = (const float*)d_in[4];   // (B,M,4)
  const float* anch  = (const float*)d_in[5];   // (A,2)
  float* out = (float*)d_out;

  // workspace layout (floats/ints, 4B aligned): ~566 KB total
  int*   winner   = (int*)d_ws;                 // NCELL ints
  float* fws      = (float*)d_ws;
  float* partialA = fws + NCELL;                // NPA
  float* corrA    = partialA + NPA;             // NBM each
  float* objA     = corrA + NBM;
  float* priorA   = objA + NBM;
  float* truA     = priorA + NBM;
  float* scoA     = truA + NBM;

  kInit<<<(NCELL + 255) / 256, 256, 0, stream>>>(winner);
  kScatter<<<(NBM + 255) / 256, 256, 0, stream>>>(tobj, winner);
  kConfSum<<<NPA, 256, 0, stream>>>(pred, partialA);
  kObj<<<NBM / 8, 256, 0, stream>>>(cls, pred, tlab, tobj, anch, winner,
                                    corrA, objA, priorA, truA, scoA);
  kFinal<<<1, 256, 0, stream>>>(partialA, corrA, objA, priorA, truA, scoA, iter, out);
}